// WPointCloudLoss_44444321579529
// MI455X (gfx1250) — compile-verified
//
#include <hip/hip_runtime.h>
#include <hip/hip_bf16.h>

#define EPS_F     0.1f
#define INV_EPS_F 10.0f
#define BIG_F     1e20f
#define HUGE_F    1e30f
#define SMALL_F   1e-7f
#define NITER_I   20
#define NB        32
#define NN        1024
#define DXY       (1.0f/1024.0f)

typedef __attribute__((ext_vector_type(2))) float v2f;
typedef __attribute__((ext_vector_type(8))) float v8f;
typedef int gv4i __attribute__((vector_size(16)));   // matches builtin param

#define ASG __attribute__((address_space(1)))
#define ASL __attribute__((address_space(3)))

#if defined(__has_builtin)
#if __has_builtin(__builtin_amdgcn_wmma_f32_16x16x4_f32)
#define USE_WMMA_F32 1
#endif
#if __has_builtin(__builtin_amdgcn_global_load_async_to_lds_b128)
#define USE_ASYNC_LDS 1
#endif
#endif

__device__ __forceinline__ void wait_async0() {
#if defined(__has_builtin)
#if __has_builtin(__builtin_amdgcn_s_wait_asynccnt)
  __builtin_amdgcn_s_wait_asynccnt(0);
#else
  asm volatile("s_wait_asynccnt 0x0" ::: "memory");
#endif
#else
  asm volatile("s_wait_asynccnt 0x0" ::: "memory");
#endif
}

// ---------------- workspace layout (float offsets) ----------------
#define OFF_U      ((size_t)0)
#define OFF_V      (OFF_U + NB*NN)
#define OFF_B      (OFF_V + NB*NN)
#define OFF_A      (OFF_B + NB*NN)
#define OFF_BNEW   (OFF_A + NB*NN)
#define OFF_S2     (OFF_BNEW + NB*NN)
#define OFF_ROWSUM (OFF_S2 + NB*NN)
#define OFF_COLSUM (OFF_ROWSUM + NB*NN)
#define OFF_TRANS  (OFF_COLSUM + NB*NN)   // NB floats
#define OFF_MAXA   (OFF_TRANS + NB)       // 1 uint slot
#define OFF_MAXB   (OFF_MAXA + 1)         // 1 uint slot

__global__ __launch_bounds__(256) void k_init(float* __restrict__ ws) {
  int idx = blockIdx.x * 256 + threadIdx.x;
  if (idx < NB * NN) {
    ws[OFF_U + idx] = 0.0f;
    ws[OFF_V + idx] = 0.0f;
    ws[OFF_B + idx] = 1.0f;
    ws[OFF_COLSUM + idx] = 0.0f;
  }
  if (idx < NB) ws[OFF_TRANS + idx] = 0.0f;
}

__global__ __launch_bounds__(256) void k_zero(float* __restrict__ ws) {
  int idx = blockIdx.x * 256 + threadIdx.x;
  if (idx < NB * NN) ws[OFF_S2 + idx] = 0.0f;
  if (idx < 2) ((unsigned*)(ws + OFF_MAXA))[idx] = 0u;
}

// ---- row pass: s_i = dy * sum_j K_ij b_j ; a_i = clamp(1/s_i); track max(a)
// One block = one (batch, 16-row tile). C tiles are double-buffered into LDS
// via GLOBAL_LOAD_ASYNC_TO_LDS_B128 (ASYNCcnt); 8 waves each consume a 16-col
// slice, computing K*b on the fly into 16x4 A-fragments that feed
// V_WMMA_F32_16X16X4_F32 with an all-ones B (exact f32 row-sum reduction).
__global__ __launch_bounds__(256) void k_row(const float* __restrict__ C,
                                             float* __restrict__ ws) {
  const int batch = blockIdx.y;
  const int r0 = blockIdx.x * 16;
  const float* __restrict__ Cb = C + (size_t)batch * NN * NN;
  const float* __restrict__ u  = ws + OFF_U + (size_t)batch * NN;
  const float* __restrict__ v  = ws + OFF_V + (size_t)batch * NN;
  const float* __restrict__ bv = ws + OFF_B + (size_t)batch * NN;
  float* __restrict__ a        = ws + OFF_A + (size_t)batch * NN;
  unsigned* maxa = (unsigned*)(ws + OFF_MAXA);

  __shared__ float srow[16];
  const int t = threadIdx.x;
  const int lane = t & 31;
  const int wave = t >> 5;
  if (t < 16) srow[t] = 0.0f;

#if defined(USE_WMMA_F32) && defined(USE_ASYNC_LDS)
  __shared__ float ct[2][16][132];   // raw C tiles, padded rows (16B-mult stride)
  v8f acc = {};
  v2f Bones; Bones.x = 1.0f; Bones.y = 1.0f;
  const float uL = u[r0 + (lane & 15)];       // per-lane row is loop-invariant

  // prefetch chunk 0 into buffer 0 (each thread copies 2x16B, coalesced)
#pragma unroll
  for (int e = 0; e < 2; ++e) {
    int idx = t + 256 * e;              // 0..511
    int r = idx >> 5;                   // 0..15
    int seg = (idx & 31) * 4;           // float offset within row, 16B units
    __builtin_amdgcn_global_load_async_to_lds_b128(
        (ASG gv4i*)(Cb + (size_t)(r0 + r) * NN + seg),
        (ASL gv4i*)&ct[0][r][seg], 0, 0);
  }

  int buf = 0;
  for (int c0 = 0; c0 < NN; c0 += 128) {
    wait_async0();        // my async loads (current chunk) have landed
    __syncthreads();      // everyone's landed; prior consumers of buf^1 done
    if (c0 + 128 < NN) {  // prefetch next chunk into the other buffer
#pragma unroll
      for (int e = 0; e < 2; ++e) {
        int idx = t + 256 * e;
        int r = idx >> 5;
        int seg = (idx & 31) * 4;
        __builtin_amdgcn_global_load_async_to_lds_b128(
            (ASG gv4i*)(Cb + (size_t)(r0 + r) * NN + c0 + 128 + seg),
            (ASL gv4i*)&ct[buf ^ 1][r][seg], 0, 0);
      }
    }
    const int cb = wave * 16;
#pragma unroll
    for (int p = 0; p < 4; ++p) {
      // A fragment (16x4 f32): lane L -> row L%16, cols cb+4p+2*(L/16)+{0,1}
      int rr = lane & 15;
      int ccol = cb + 4 * p + 2 * (lane >> 4);
      float cA = ct[buf][rr][ccol];
      float cB = ct[buf][rr][ccol + 1];
      int g = c0 + ccol;
      v2f A;
      A.x = fminf(__expf((uL + v[g]     - cA) * INV_EPS_F), HUGE_F) * bv[g];
      A.y = fminf(__expf((uL + v[g + 1] - cB) * INV_EPS_F), HUGE_F) * bv[g + 1];
      // D[i,n] += sum_q A[i,q] * 1 -> exact row partials, replicated over n
      acc = __builtin_amdgcn_wmma_f32_16x16x4_f32(
          false, A, false, Bones, (short)0, acc, false, false);
    }
    buf ^= 1;
  }

  // D layout: VGPR j, lanes 0-15: (M=j, N=lane); lanes 16-31: (M=j+8, N=lane-16)
  if (lane == 0) {
#pragma unroll
    for (int j = 0; j < 8; ++j) atomicAdd(&srow[j], acc[j]);
  } else if (lane == 16) {
#pragma unroll
    for (int j = 0; j < 8; ++j) atomicAdd(&srow[8 + j], acc[j]);
  }

#else  // -------- fallback: synchronous staging --------
  __shared__ float w[16][132];
#ifdef USE_WMMA_F32
  v8f acc = {};
  v2f Bones; Bones.x = 1.0f; Bones.y = 1.0f;
#else
  float fb = 0.0f;
#endif
  for (int c0 = 0; c0 < NN; c0 += 128) {
#pragma unroll
    for (int e = 0; e < 8; ++e) {
      int idx = t + 256 * e;
      int r  = idx >> 7;
      int cc = idx & 127;
      int col = c0 + cc;
      float cval = Cb[(size_t)(r0 + r) * NN + col];
      float kv = fminf(__expf((u[r0 + r] + v[col] - cval) * INV_EPS_F), HUGE_F);
      w[r][cc] = kv * bv[col];
    }
    __syncthreads();
    const int cb = wave * 16;
#pragma unroll
    for (int p = 0; p < 4; ++p) {
      int rr = lane & 15;
      int ccol = cb + 4 * p + 2 * (lane >> 4);
      v2f A;
      A.x = w[rr][ccol];
      A.y = w[rr][ccol + 1];
#ifdef USE_WMMA_F32
      acc = __builtin_amdgcn_wmma_f32_16x16x4_f32(
          false, A, false, Bones, (short)0, acc, false, false);
#else
      fb += A.x + A.y;
#endif
    }
    __syncthreads();
  }
#ifdef USE_WMMA_F32
  if (lane == 0) {
#pragma unroll
    for (int j = 0; j < 8; ++j) atomicAdd(&srow[j], acc[j]);
  } else if (lane == 16) {
#pragma unroll
    for (int j = 0; j < 8; ++j) atomicAdd(&srow[8 + j], acc[j]);
  }
#else
  float tmp = fb + __shfl_xor(fb, 16);
  if (lane < 16) atomicAdd(&srow[lane], tmp);
#endif
#endif // async/sync

  __syncthreads();
  if (t < 16) {
    float s = srow[t] * DXY;                 // dy
    float av = fminf(1.0f / s, HUGE_F);      // clamp(mu/s, 0, HUGE), mu=1
    a[r0 + t] = av;
    atomicMax(maxa, __float_as_uint(av));    // positive floats: bits monotone
  }
}

// ---- col pass: s2_j += sum_i K_ij a_i  (lane = column, fully coalesced)
__global__ __launch_bounds__(256) void k_col(const float* __restrict__ C,
                                             float* __restrict__ ws) {
  const int batch = blockIdx.z;
  const int c0 = blockIdx.x * 256;
  const int r0 = blockIdx.y * 256;
  const float* __restrict__ Cb = C + (size_t)batch * NN * NN;
  const float* __restrict__ u  = ws + OFF_U + (size_t)batch * NN;
  const float* __restrict__ v  = ws + OFF_V + (size_t)batch * NN;
  const float* __restrict__ a  = ws + OFF_A + (size_t)batch * NN;
  float* __restrict__ s2       = ws + OFF_S2 + (size_t)batch * NN;

  const int col = c0 + threadIdx.x;
  const float vv = v[col];
  float acc = 0.0f;
  for (int r = r0; r < r0 + 256; ++r) {
    float cval = Cb[(size_t)r * NN + col];
    float kv = fminf(__expf((u[r] + vv - cval) * INV_EPS_F), HUGE_F);
    acc = fmaf(kv, a[r], acc);
  }
  atomicAdd(&s2[col], acc);
}

__global__ __launch_bounds__(256) void k_bnew(float* __restrict__ ws) {
  int idx = blockIdx.x * 256 + threadIdx.x;   // 0..32767
  float s2 = ws[OFF_S2 + idx] * DXY;          // dx
  float bn = fminf(1.0f / s2, HUGE_F);        // clamp(nu/s2), nu=1
  ws[OFF_BNEW + idx] = bn;
  atomicMax((unsigned*)(ws + OFF_MAXB), __float_as_uint(bn));
}

__global__ __launch_bounds__(256) void k_update(float* __restrict__ ws, int ii) {
  int idx = blockIdx.x * 256 + threadIdx.x;
  float ma = __uint_as_float(((unsigned*)(ws + OFF_MAXA))[0]);
  float mb = __uint_as_float(((unsigned*)(ws + OFF_MAXB))[0]);
  bool cond = (ma > BIG_F) || (mb > BIG_F) || (ii == NITER_I - 1);
  float bn = ws[OFF_BNEW + idx];
  if (cond) {                                 // absorb potentials
    ws[OFF_U + idx] += EPS_F * __logf(ws[OFF_A + idx]);
    ws[OFF_V + idx] += EPS_F * __logf(bn);
    ws[OFF_B + idx] = 1.0f;
  } else {
    ws[OFF_B + idx] = bn;
  }
}

// ---- finals: write K, row marginals, transport
__global__ __launch_bounds__(256) void k_final_rows(const float* __restrict__ C,
                                                    float* __restrict__ ws,
                                                    float* __restrict__ out) {
  const int batch = blockIdx.y;
  const int r0 = blockIdx.x * 64;
  const int lane = threadIdx.x & 31;
  const int wave = threadIdx.x >> 5;
  const float* __restrict__ Cb = C + (size_t)batch * NN * NN;
  const float* __restrict__ u  = ws + OFF_U + (size_t)batch * NN;
  const float* __restrict__ v  = ws + OFF_V + (size_t)batch * NN;
  float* __restrict__ Kout = out + 96 + (size_t)batch * NN * NN;
  float* __restrict__ rowsum = ws + OFF_ROWSUM + (size_t)batch * NN;

  float tpart = 0.0f;
  for (int k = 0; k < 8; ++k) {
    int row = r0 + wave * 8 + k;
    float ui = u[row];
    float rtot = 0.0f;
    for (int c = lane; c < NN; c += 32) {
      float cval = Cb[(size_t)row * NN + c];
      float kv = fminf(__expf((ui + v[c] - cval) * INV_EPS_F), HUGE_F);
      Kout[(size_t)row * NN + c] = kv;
      rtot += kv;
      tpart = fmaf(kv, cval, tpart);
    }
    for (int off = 16; off > 0; off >>= 1) rtot += __shfl_xor(rtot, off);
    if (lane == 0) rowsum[row] = rtot;
  }
  for (int off = 16; off > 0; off >>= 1) tpart += __shfl_xor(tpart, off);
  if (lane == 0) atomicAdd(ws + OFF_TRANS + batch, tpart * DXY * DXY);
}

__global__ __launch_bounds__(256) void k_final_cols(const float* __restrict__ Kall,
                                                    float* __restrict__ ws) {
  const int batch = blockIdx.z;
  const int c0 = blockIdx.x * 256;
  const int r0 = blockIdx.y * 256;
  const float* __restrict__ Kb = Kall + (size_t)batch * NN * NN;
  const int col = c0 + threadIdx.x;
  float acc = 0.0f;
  for (int r = r0; r < r0 + 256; ++r) acc += Kb[(size_t)r * NN + col];
  atomicAdd(ws + OFF_COLSUM + (size_t)batch * NN + col, acc);
}

__device__ __forceinline__ float klterm(float x) {
  float div = x / (1.0f + SMALL_F);                    // y = 1
  return div * logf(div + SMALL_F) - div + 1.0f;
}

__global__ __launch_bounds__(256) void k_final_out(float* __restrict__ ws,
                                                   float* __restrict__ out) {
  const int batch = blockIdx.x;
  const int t = threadIdx.x;
  __shared__ float red[256];
  float c1 = 0.0f, c2 = 0.0f;
  for (int i = t; i < NN; i += 256) {
    c1 += klterm(ws[OFF_ROWSUM + (size_t)batch * NN + i] * DXY);
    c2 += klterm(ws[OFF_COLSUM + (size_t)batch * NN + i] * DXY);
  }
  red[t] = c1 * DXY + c2 * DXY;   // cons1*dx + cons2*dy
  __syncthreads();
  for (int s = 128; s > 0; s >>= 1) {
    if (t < s) red[t] += red[t + s];
    __syncthreads();
  }
  if (t == 0) {
    float cons = red[0];
    float tr = ws[OFF_TRANS + batch];
    out[batch]      = cons + tr;   // p_opt
    out[32 + batch] = tr;          // transport
    out[64 + batch] = cons;        // constrain
  }
}

extern "C" void kernel_launch(void* const* d_in, const int* in_sizes, int n_in,
                              void* d_out, int out_size, void* d_ws, size_t ws_size,
                              hipStream_t stream) {
  const float* C = (const float*)d_in[0];
  float* out = (float*)d_out;
  float* ws = (float*)d_ws;

  k_init<<<dim3((NB * NN + 255) / 256), 256, 0, stream>>>(ws);

  for (int ii = 0; ii < NITER_I; ++ii) {
    k_zero<<<dim3(128), 256, 0, stream>>>(ws);
    k_row<<<dim3(64, 32), 256, 0, stream>>>(C, ws);
    k_col<<<dim3(4, 4, 32), 256, 0, stream>>>(C, ws);
    k_bnew<<<dim3(128), 256, 0, stream>>>(ws);
    k_update<<<dim3(128), 256, 0, stream>>>(ws, ii);
  }

  k_final_rows<<<dim3(16, 32), 256, 0, stream>>>(C, ws, out);
  k_final_cols<<<dim3(4, 4, 32), 256, 0, stream>>>(out + 96, ws);
  k_final_out<<<dim3(32), 256, 0, stream>>>(ws, out);
}